// PixelwiseNet_78383153152428
// MI455X (gfx1250) — compile-verified
//
#include <hip/hip_runtime.h>
#include <hip/hip_bf16.h>
#include <math.h>

// fp32 WMMA operand types (16x16x4: A/B = 2 VGPRs, C/D = 8 VGPRs per lane, wave32)
typedef float v2f __attribute__((ext_vector_type(2)));
typedef float v8f __attribute__((ext_vector_type(8)));

#define BN_EPS 1e-5f

// Problem constants from setup_inputs(): [B,G,D,H,W] = [4,8,48,128,160]
#define NB   4
#define NC   8
#define ND   48
#define NH   128
#define NW   160
#define NHW  (NH * NW)          /* 20480 */
#define NCS  (ND * NHW)         /* channel stride: 983040 */

__global__ __launch_bounds__(256)
void pixelwise_net_fused(const float* __restrict__ x1,
                         const float* __restrict__ w0,
                         const float* __restrict__ g0, const float* __restrict__ bb0,
                         const float* __restrict__ m0, const float* __restrict__ v0,
                         const float* __restrict__ w1,
                         const float* __restrict__ g1, const float* __restrict__ bb1,
                         const float* __restrict__ m1, const float* __restrict__ v1,
                         const float* __restrict__ w2, const float* __restrict__ b2,
                         float* __restrict__ out)
{
    const int lane = threadIdx.x & 31;
    const int l16  = lane & 15;
    const int hi   = lane >> 4;           // 0 = low half-wave, 1 = high half-wave
    const bool lo  = (hi == 0);

    // One wave = 32 consecutive (h,w) pixels = two 16-voxel WMMA tiles.
    const int gw          = (blockIdx.x * blockDim.x + threadIdx.x) >> 5;
    const int wavesPerImg = NHW / 32;     // 640
    const int b           = gw / wavesPerImg;
    const int hwb         = (gw - b * wavesPerImg) * 32;

    // ---- conv0 A operand: w0 [16 out x 8 in], two K=4 slices -------------
    // A layout 16x4: lanes 0-15 hold K=0(V0),K=1(V1); lanes 16-31 K=2,K=3.
    float a0[2], a1[2];
#pragma unroll
    for (int p = 0; p < 2; ++p) {
        a0[p] = w0[l16 * 8 + 4 * p + 0 + 2 * hi];
        a1[p] = w0[l16 * 8 + 4 * p + 1 + 2 * hi];
    }

    // ---- conv1 A operand: w1 [8x16] zero-padded to 16x16, four K=4 slices -
    float wa0[4], wa1[4];
#pragma unroll
    for (int p = 0; p < 4; ++p) {
        wa0[p] = (l16 < 8) ? w1[l16 * 16 + 4 * p + 0 + 2 * hi] : 0.0f;
        wa1[p] = (l16 < 8) ? w1[l16 * 16 + 4 * p + 1 + 2 * hi] : 0.0f;
    }

    // ---- BN0 folded scale/shift; conv0 D layout: reg r holds ch r (lo) / r+8 (hi)
    float sc0[8], sh0[8];
#pragma unroll
    for (int r = 0; r < 8; ++r) {
        const int ch = r + 8 * hi;
        const float inv = rsqrtf(v0[ch] + BN_EPS);
        sc0[r] = g0[ch] * inv;
        sh0[r] = bb0[ch] - m0[ch] * sc0[r];
    }

    // ---- BN1 folded (channel == reg index on low lanes) + conv2 weights ---
    float sc1[8], sh1[8], w2r[8];
#pragma unroll
    for (int r = 0; r < 8; ++r) {
        const float inv = rsqrtf(v1[r] + BN_EPS);
        sc1[r] = g1[r] * inv;
        sh1[r] = bb1[r] - m1[r] * sc1[r];
        w2r[r] = w2[r];
    }
    const float bias2 = b2[0];

    // Per-lane base pointer into x1 for this wave's 32 pixels.
    const float* xb = x1 + (size_t)b * NC * NCS + hwb + l16;

    float vmax[2] = {0.0f, 0.0f};         // sigmoid() > 0, so 0 is a safe identity

    for (int d = 0; d < ND; ++d) {
        const float* xd = xb + (size_t)d * NHW;
#pragma unroll
        for (int t = 0; t < 2; ++t) {
            const float* xt = xd + 16 * t;

            // ---- conv0: D0[16 outch x 16 vox] = w0 * x, K=8 in two WMMAs --
            v8f acc = {};
#pragma unroll
            for (int p = 0; p < 2; ++p) {
                v2f A; A[0] = a0[p]; A[1] = a1[p];
                // B layout 4x16: V0 = {K=0 lo, K=2 hi}, V1 = {K=1 lo, K=3 hi}
                v2f B;
                B[0] = xt[(size_t)(4 * p + 0 + 2 * hi) * NCS];
                B[1] = xt[(size_t)(4 * p + 1 + 2 * hi) * NCS];
                acc = __builtin_amdgcn_wmma_f32_16x16x4_f32(
                          false, A, false, B, (short)0, acc, false, false);
            }

            // ---- BN0 + ReLU (per-reg channel scale) -----------------------
            float h[8], sw[8];
#pragma unroll
            for (int r = 0; r < 8; ++r)
                h[r] = fmaxf(fmaf(acc[r], sc0[r], sh0[r]), 0.0f);
            // cross-half re-striping: D rows {r lo, r+8 hi} -> B slices
#pragma unroll
            for (int r = 0; r < 8; ++r)
                sw[r] = __shfl_xor(h[r], 16, 32);

            // ---- conv1: D1[16(8)ch x 16 vox] = w1pad * h, K=16 in 4 WMMAs -
            v8f acc1 = {};
            {
                v2f A, B;
                A[0] = wa0[0]; A[1] = wa1[0];                 // K rows 0..3
                B[0] = lo ? h[0] : sw[2];  B[1] = lo ? h[1] : sw[3];
                acc1 = __builtin_amdgcn_wmma_f32_16x16x4_f32(
                           false, A, false, B, (short)0, acc1, false, false);

                A[0] = wa0[1]; A[1] = wa1[1];                 // K rows 4..7
                B[0] = lo ? h[4] : sw[6];  B[1] = lo ? h[5] : sw[7];
                acc1 = __builtin_amdgcn_wmma_f32_16x16x4_f32(
                           false, A, false, B, (short)0, acc1, false, false);

                A[0] = wa0[2]; A[1] = wa1[2];                 // K rows 8..11
                B[0] = lo ? sw[0] : h[2];  B[1] = lo ? sw[1] : h[3];
                acc1 = __builtin_amdgcn_wmma_f32_16x16x4_f32(
                           false, A, false, B, (short)0, acc1, false, false);

                A[0] = wa0[3]; A[1] = wa1[3];                 // K rows 12..15
                B[0] = lo ? sw[4] : h[6];  B[1] = lo ? sw[5] : h[7];
                acc1 = __builtin_amdgcn_wmma_f32_16x16x4_f32(
                           false, A, false, B, (short)0, acc1, false, false);
            }

            // ---- BN1 + ReLU + conv2 dot + bias + sigmoid + depth-max ------
            // (valid on low lanes: D1 reg r, lanes 0-15 = (ch r, vox lane))
            float z = bias2;
#pragma unroll
            for (int r = 0; r < 8; ++r)
                z = fmaf(w2r[r], fmaxf(fmaf(acc1[r], sc1[r], sh1[r]), 0.0f), z);
            const float sig = 1.0f / (1.0f + __expf(-z));
            vmax[t] = fmaxf(vmax[t], sig);
        }
    }

    // Tile A result lives in low lanes, tile B in low lanes too: move tile B
    // to the high half so the store is one fully coalesced 128B write.
    const float swb = __shfl_xor(vmax[1], 16, 32);
    const float res = lo ? vmax[0] : swb;
    out[(size_t)b * NHW + hwb + lane] = res;
}

extern "C" void kernel_launch(void* const* d_in, const int* in_sizes, int n_in,
                              void* d_out, int out_size, void* d_ws, size_t ws_size,
                              hipStream_t stream) {
    (void)in_sizes; (void)n_in; (void)d_ws; (void)ws_size; (void)out_size;
    const float* x1 = (const float*)d_in[0];
    const float* w0 = (const float*)d_in[1];
    const float* g0 = (const float*)d_in[2];
    const float* b0 = (const float*)d_in[3];
    const float* m0 = (const float*)d_in[4];
    const float* v0 = (const float*)d_in[5];
    const float* w1 = (const float*)d_in[6];
    const float* g1 = (const float*)d_in[7];
    const float* b1 = (const float*)d_in[8];
    const float* m1 = (const float*)d_in[9];
    const float* v1 = (const float*)d_in[10];
    const float* w2 = (const float*)d_in[11];
    const float* b2 = (const float*)d_in[12];
    float* out = (float*)d_out;

    // 4*20480 output pixels, 32 per wave -> 2560 waves -> 320 blocks of 256.
    const int totalThreads = (NB * NHW / 32) * 32;
    const int block = 256;
    const int grid  = totalThreads / block;
    pixelwise_net_fused<<<grid, block, 0, stream>>>(
        x1, w0, g0, b0, m0, v0, w1, g1, b1, m1, v1, w2, b2, out);
}